// SSIMLoss_61452392071727
// MI455X (gfx1250) — compile-verified
//
#include <hip/hip_runtime.h>

typedef __attribute__((ext_vector_type(2))) float v2f;
typedef __attribute__((ext_vector_type(8))) float v8f;

#define IMG   512
#define NPLN  124          // 4 * 31 planes
#define TILE  64           // output tile edge
#define HALO  5
#define XN    (TILE + 2*HALO)   // 74 valid halo extent
#define XR    80           // padded LDS input rows (5 x 16)
#define XP    76           // LDS input pitch (floats, even)
#define HR    80           // intermediate rows
#define HP    66           // intermediate pitch
#define RB    5            // stage-1 row blocks
#define CB    4            // stage-1 col blocks
#define C1V   1.0e-4f      // 0.01^2
#define C2V   9.0e-4f      // 0.03^2
#define NTOT  32505856.0   // 4*31*512*512

#if defined(__HIP_DEVICE_COMPILE__) && \
    __has_builtin(__builtin_amdgcn_global_load_async_to_lds_b32) && \
    __has_builtin(__builtin_amdgcn_s_wait_asynccnt)
#define USE_ASYNC 1
typedef __attribute__((address_space(1))) int g_i32;   // global (AS1) word
typedef __attribute__((address_space(3))) int l_i32;   // LDS (AS3) word
#else
#define USE_ASYNC 0
#endif

// Normalized 11-tap Gaussian, sigma = 1.5
__device__ __constant__ float GW[11] = {
    0.00102838f, 0.00759876f, 0.03600077f, 0.10936069f, 0.21300553f,
    0.26601172f,
    0.21300553f, 0.10936069f, 0.03600077f, 0.00759876f, 0.00102838f};

// Banded filter matrix element: W[k][n] = g[k-n] if 0 <= k-n <= 10 else 0
__device__ __forceinline__ float bandw(int k, int n) {
    const int d = k - n;
    return (d >= 0 && d < 11) ? GW[d] : 0.0f;
}

extern "C" __global__ void __launch_bounds__(256)
ssim_tile_kernel(const float* __restrict__ pred, const float* __restrict__ tgt,
                 double* __restrict__ accum)
{
    __shared__ float sX[2][XR * XP];   // zero-padded pred / target halo tile
    __shared__ float sH[5][HR * HP];   // horizontally-filtered p, t, p2, t2, pt
    __shared__ float sRed[8];

    const int tid  = threadIdx.x;
    const int wave = __builtin_amdgcn_readfirstlane(tid >> 5);  // uniform per wave
    const int lane = tid & 31;
    const int lh   = lane >> 4;        // lane half (K-pair select)
    const int lm   = lane & 15;        // lane within half

    const int tpr = IMG / TILE;        // 8
    const int tpp = tpr * tpr;         // 64
    const int bid = blockIdx.x;
    const int pln = bid / tpp;
    const int rem = bid - pln * tpp;
    const int ty  = rem / tpr;
    const int tx  = rem - ty * tpr;
    const int y0  = ty * TILE - HALO;
    const int x0  = tx * TILE - HALO;
    const size_t pb = (size_t)pln * IMG * IMG;

    // Per-lane banded Gaussian fragment, one K-pair per chunk.  These are the
    // stage-1 B-matrix and the stage-2 A-matrix (identical banded values).
    float bwa[7], bwb[7];
#pragma unroll
    for (int kk = 0; kk < 7; ++kk) {
        const int kb = kk * 4 + lh * 2;
        bwa[kk] = bandw(kb, lm);
        bwb[kk] = bandw(kb + 1, lm);
    }

    // ---- stage 0: global -> LDS ----
#if USE_ASYNC
    // Interior tiles: the 74x74 valid window is fully in-bounds.  Pad rows /
    // cols feed only zero rows of the banded matrix but must hold finite
    // values, so zero them (disjoint LDS addresses -> no ordering hazard
    // with the async writes), then bulk-issue async LDS loads.
    if (ty >= 1 && ty <= 6 && tx >= 1 && tx <= 6) {
        for (int i = tid; i < XR * XP; i += 256) {
            const int r = i / XP;
            const int c = i - r * XP;
            if (r >= XN || c >= XN) { sX[0][i] = 0.0f; sX[1][i] = 0.0f; }
        }
        for (int i = tid; i < XN * XN; i += 256) {
            const int r  = i / XN;
            const int c  = i - r * XN;
            const int li = r * XP + c;
            const size_t o = pb + (size_t)(y0 + r) * IMG + (x0 + c);
            __builtin_amdgcn_global_load_async_to_lds_b32(
                (g_i32*)(size_t)(pred + o), (l_i32*)&sX[0][li], 0, 0);
            __builtin_amdgcn_global_load_async_to_lds_b32(
                (g_i32*)(size_t)(tgt + o), (l_i32*)&sX[1][li], 0, 0);
        }
        __builtin_amdgcn_s_wait_asynccnt(0);
    } else
#endif
    {
        for (int i = tid; i < XR * XP; i += 256) {
            const int r  = i / XP;
            const int c  = i - r * XP;
            const int gy = y0 + r;
            const int gx = x0 + c;
            float pv = 0.0f, tv = 0.0f;
            if (r < XN && c < XN && gy >= 0 && gy < IMG && gx >= 0 && gx < IMG) {
                const size_t o = pb + (size_t)gy * IMG + gx;
                pv = pred[o];
                tv = tgt[o];
            }
            sX[0][i] = pv;
            sX[1][i] = tv;
        }
    }
    __syncthreads();

    // ---- stage 1: horizontal Gaussian as H(16x16) = A(16x28) * B(28x16) ----
    // One subtile per task; all 5 quantities share the same LDS loads and the
    // same banded B fragment (A for p2/t2/pt formed with 3 register muls).
    for (int sub = wave; sub < RB * CB; sub += 8) {
        const int r0 = __builtin_amdgcn_readfirstlane((sub / CB) * 16);
        const int c0 = __builtin_amdgcn_readfirstlane((sub % CB) * 16);
        const v8f zero = {0.f, 0.f, 0.f, 0.f, 0.f, 0.f, 0.f, 0.f};
        v8f acc[5] = {zero, zero, zero, zero, zero};
#pragma unroll
        for (int kk = 0; kk < 7; ++kk) {
            const int kb  = kk * 4 + lh * 2;       // this lane's K pair
            const int off = (r0 + lm) * XP + c0 + kb;
            const float p0 = sX[0][off];
            const float p1 = sX[0][off + 1];
            const float t0 = sX[1][off];
            const float t1 = sX[1][off + 1];
            const v2f B  = {bwa[kk], bwb[kk]};
            const v2f Ap  = {p0,      p1};
            const v2f At  = {t0,      t1};
            const v2f App = {p0 * p0, p1 * p1};
            const v2f Att = {t0 * t0, t1 * t1};
            const v2f Apt = {p0 * t0, p1 * t1};
            acc[0] = __builtin_amdgcn_wmma_f32_16x16x4_f32(false, Ap,  false, B, (short)0, acc[0], false, false);
            acc[1] = __builtin_amdgcn_wmma_f32_16x16x4_f32(false, At,  false, B, (short)0, acc[1], false, false);
            acc[2] = __builtin_amdgcn_wmma_f32_16x16x4_f32(false, App, false, B, (short)0, acc[2], false, false);
            acc[3] = __builtin_amdgcn_wmma_f32_16x16x4_f32(false, Att, false, B, (short)0, acc[3], false, false);
            acc[4] = __builtin_amdgcn_wmma_f32_16x16x4_f32(false, Apt, false, B, (short)0, acc[4], false, false);
        }
        // C/D layout: VGPR e -> M = e (lanes 0-15) / e+8 (lanes 16-31), N = lm
#pragma unroll
        for (int q = 0; q < 5; ++q) {
#pragma unroll
            for (int e = 0; e < 8; ++e) {
                const int m = r0 + e + lh * 8;
                sH[q][m * HP + c0 + lm] = acc[q][e];
            }
        }
    }
    __syncthreads();

    // ---- stage 2: vertical Gaussian V(16x16) = G(16x28) * H(28x16) + SSIM ----
    float lsum = 0.0f;
    for (int st = wave; st < 16; st += 8) {
        const int r0 = __builtin_amdgcn_readfirstlane((st >> 2) * 16);
        const int c0 = __builtin_amdgcn_readfirstlane((st & 3) * 16);
        const v8f zero = {0.f, 0.f, 0.f, 0.f, 0.f, 0.f, 0.f, 0.f};
        v8f macc[5] = {zero, zero, zero, zero, zero};
#pragma unroll
        for (int kk = 0; kk < 7; ++kk) {
            const int kb = kk * 4 + lh * 2;
            const v2f A = {bwa[kk], bwb[kk]};       // G[m][k] = g[k-m]
            const int h0 = (r0 + kb) * HP + c0 + lm;
#pragma unroll
            for (int q = 0; q < 5; ++q) {
                const v2f B = {sH[q][h0], sH[q][h0 + HP]};
                macc[q] = __builtin_amdgcn_wmma_f32_16x16x4_f32(
                    false, A, false, B, (short)0, macc[q], false, false);
            }
        }
        // elementwise SSIM map for this subtile
#pragma unroll
        for (int e = 0; e < 8; ++e) {
            const float mp  = macc[0][e];
            const float mt  = macc[1][e];
            const float ep2 = macc[2][e];
            const float et2 = macc[3][e];
            const float ept = macc[4][e];
            const float mp2 = mp * mp;
            const float mt2 = mt * mt;
            const float mc  = mp * mt;
            const float sp  = fmaxf(ep2 - mp2, 0.0f);
            const float stv = fmaxf(et2 - mt2, 0.0f);
            const float sc  = ept - mc;
            const float num = (2.0f * mc + C1V) * (2.0f * sc + C2V);
            const float den = (mp2 + mt2 + C1V) * (sp + stv + C2V);
            lsum += num / den;
        }
    }

    // ---- reduction: lanes -> waves -> block -> global double accumulator ----
#pragma unroll
    for (int off = 16; off > 0; off >>= 1)
        lsum += __shfl_down(lsum, off, 32);
    if (lane == 0) sRed[wave] = lsum;
    __syncthreads();
    if (tid == 0) {
        float bs = 0.0f;
#pragma unroll
        for (int w = 0; w < 8; ++w) bs += sRed[w];
        atomicAdd(accum, (double)bs);
    }
}

extern "C" __global__ void ssim_init_kernel(double* accum) {
    if (threadIdx.x == 0) *accum = 0.0;
}

extern "C" __global__ void ssim_final_kernel(const double* __restrict__ accum,
                                             float* __restrict__ out) {
    if (threadIdx.x == 0)
        out[0] = 1.0f - (float)(*accum / NTOT);
}

extern "C" void kernel_launch(void* const* d_in, const int* in_sizes, int n_in,
                              void* d_out, int out_size, void* d_ws, size_t ws_size,
                              hipStream_t stream) {
    const float* pred = (const float*)d_in[0];
    const float* tgt  = (const float*)d_in[1];
    float*  out   = (float*)d_out;
    double* accum = (double*)d_ws;

    hipLaunchKernelGGL(ssim_init_kernel, dim3(1), dim3(32), 0, stream, accum);

    const int nblocks = NPLN * (IMG / TILE) * (IMG / TILE);  // 124 * 64 = 7936
    hipLaunchKernelGGL(ssim_tile_kernel, dim3(nblocks), dim3(256), 0, stream,
                       pred, tgt, accum);

    hipLaunchKernelGGL(ssim_final_kernel, dim3(1), dim3(32), 0, stream,
                       accum, out);
}